// GCN3_16226386444397
// MI455X (gfx1250) — compile-verified
//
#include <hip/hip_runtime.h>
#include <hip/hip_bf16.h>

#define H_DIM 128
#define BN_EPS 1e-5f
#define WPAD 136   // LDS row pitch in dwords: 128 + 8 (TDM pad) -> conflict-free ds reads

typedef float        v2f  __attribute__((ext_vector_type(2)));
typedef float        v8f  __attribute__((ext_vector_type(8)));
typedef unsigned int v4u  __attribute__((ext_vector_type(4)));
typedef int          v4i  __attribute__((ext_vector_type(4)));
typedef int          v8i  __attribute__((ext_vector_type(8)));

// ---------------- degree / normalization ----------------
__global__ void k_deg_init(float* __restrict__ deg, int n) {
    int i = blockIdx.x * blockDim.x + threadIdx.x;
    if (i < n) deg[i] = 1.0f;               // self-loop contributes weight 1
}

__global__ void k_deg_edges(const int* __restrict__ cols, const float* __restrict__ ew,
                            float* __restrict__ deg, int ne) {
    int i = blockIdx.x * blockDim.x + threadIdx.x;
    if (i < ne) atomicAdd(&deg[cols[i]], ew[i]);
}

__global__ void k_dinv(const float* __restrict__ deg, float* __restrict__ dinv, int n) {
    int i = blockIdx.x * blockDim.x + threadIdx.x;
    if (i < n) { float d = deg[i]; dinv[i] = d > 0.f ? rsqrtf(d) : 0.f; }
}

// ---------------- dense GEMM via WMMA (f32 matrix cores) ----------------
// out[N x 128] = (AFFINE ? A*scale+shift : A) @ W.
// Block = 256 thr = 8 waves; each wave owns a 16-row strip (block covers 128 rows).
// W is DMA'd once per block into LDS by the Tensor Data Mover with 8-dword row
// padding (pitch 136 dwords) so B-operand ds reads are bank-conflict-free.
// A fragments are loaded once into registers and reused across all 8 column tiles.
template<bool AFFINE>
__global__ __launch_bounds__(256) void k_gemm_wmma(
    const float* __restrict__ A, const float* __restrict__ W,
    const float* __restrict__ scale, const float* __restrict__ shift,
    float* __restrict__ out, int nrows)
{
    __shared__ float wlds[H_DIM * WPAD];    // 69,632 B

    const int lane = threadIdx.x & 31;
    const int wv   = threadIdx.x >> 5;

#if __has_builtin(__builtin_amdgcn_tensor_load_to_lds) && __has_builtin(__builtin_amdgcn_s_wait_tensorcnt)
    if (wv == 0) {
        // Build D# descriptor: 2-D tensor, 4-byte elements, 128x128 tile,
        // pad 8 dwords after every 128 dwords (one row) -> LDS pitch 136.
        typedef __attribute__((address_space(3))) float lds_f;
        const unsigned lds_addr = (unsigned)(size_t)(lds_f*)wlds;
        const unsigned long long ga = (unsigned long long)(size_t)W;
        v4u g0;
        g0[0] = 1u;                                   // count=1, user descriptor
        g0[1] = lds_addr;                             // lds_addr [63:32]
        g0[2] = (unsigned)ga;                         // global_addr [95:64]
        g0[3] = (unsigned)((ga >> 32) & 0x01FFFFFFu)  // global_addr [120:96]
              | (2u << 30);                           // type = 2 ("image")
        v8i g1;
        g1[0] = (int)((2u << 16)                      // data_size = 4 B
              |       (1u << 20)                      // pad_enable
              |       (6u << 22)                      // pad_interval: 2^(6+1)=128 dwords
              |       (7u << 25));                    // pad_amount: 7+1 = 8 dwords
        g1[1] = (int)(128u << 16);                    // tensor_dim0 = 128 (low16 in [63:48])
        g1[2] = (int)(128u << 16);                    // tensor_dim1 = 128 (low16 in [95:80])
        g1[3] = (int)(128u << 16);                    // tile_dim0 = 128   ([127:112])
        g1[4] = 128;                                  // tile_dim1 = 128   ([143:128])
        g1[5] = 128;                                  // tensor_dim0_stride = 128
        g1[6] = (int)(16384u << 16);                  // tensor_dim1_stride = 16384 (low16)
        g1[7] = 0;
        v4i g2 = {0, 0, 0, 0};                        // >=3-D fields unused (tile_dim2 = 0)
        v4i g3 = {0, 0, 0, 0};
        v8i g4 = {0, 0, 0, 0, 0, 0, 0, 0};            // extension group (unused, zero)
        __builtin_amdgcn_tensor_load_to_lds(g0, g1, g2, g3, g4, 0);
        __builtin_amdgcn_s_wait_tensorcnt(0);
    }
#else
    // Fallback: cooperative staging with the same padded layout.
    for (int i = threadIdx.x; i < H_DIM * H_DIM; i += 256)
        wlds[(i >> 7) * WPAD + (i & 127)] = W[i];
#endif
    __syncthreads();

    const int m16  = lane & 15;
    const int kh   = lane >> 4;                         // 0/1: K-half carried by this lane
    const int row0 = (blockIdx.x * 8 + wv) << 4;
    const int arow = min(row0 + m16, nrows - 1);        // clamp: EXEC stays all-ones for WMMA
    const float* __restrict__ arp = A + (size_t)arow * H_DIM;

    // Load (and BN-affine) this wave's A strip once; reused for all 8 column tiles.
    v2f a[32];
    #pragma unroll
    for (int k0 = 0; k0 < H_DIM; k0 += 4) {
        const int ka = k0 + (kh << 1);
        v2f t;
        t[0] = arp[ka];
        t[1] = arp[ka + 1];
        if (AFFINE) {
            t[0] = t[0] * scale[ka]     + shift[ka];
            t[1] = t[1] * scale[ka + 1] + shift[ka + 1];
        }
        a[k0 >> 2] = t;
    }

    const int orow0 = row0 + (kh << 3);
    #pragma unroll 1
    for (int j = 0; j < 8; ++j) {
        const int col0 = j << 4;
        v8f c = {};
        #pragma unroll
        for (int k0 = 0; k0 < H_DIM; k0 += 4) {
            const int ka = k0 + (kh << 1);
            v2f b;
            b[0] = wlds[ka * WPAD + col0 + m16];
            b[1] = wlds[(ka + 1) * WPAD + col0 + m16];
            c = __builtin_amdgcn_wmma_f32_16x16x4_f32(false, a[k0 >> 2], false, b,
                                                      (short)0, c, false, false);
        }
        float* __restrict__ op = out + (size_t)orow0 * H_DIM + col0 + m16;
        #pragma unroll
        for (int r = 0; r < 8; ++r)
            if (orow0 + r < nrows) op[(size_t)r * H_DIM] = c[r];
    }
}

// ---------------- aggregation ----------------
// Self-loop term initializes AGG without atomics: AGG[n] = dinv[n]^2 * T[n]
__global__ void k_agg_init(const float* __restrict__ T, const float* __restrict__ dinv,
                           float* __restrict__ agg, int n) {
    int i = blockIdx.x * blockDim.x + threadIdx.x;
    if (i < n * H_DIM) {
        const int node = i >> 7;
        const float di = dinv[node];
        agg[i] = di * di * T[i];
    }
}

// One wave per edge: float4 gather of the source row (51 MB feature matrix is
// L2-resident: 192 MB L2), 4 global f32 atomics per lane into the destination row.
__global__ __launch_bounds__(256) void k_scatter(
    const int* __restrict__ rows, const int* __restrict__ cols,
    const float* __restrict__ ew, const float* __restrict__ dinv,
    const float* __restrict__ T, float* __restrict__ agg, int ne)
{
    const int e = blockIdx.x * 8 + (threadIdx.x >> 5);
    if (e >= ne) return;
    const int lane  = threadIdx.x & 31;
    const int r     = rows[e];
    const int cnode = cols[e];
    const float nrm = dinv[r] * ew[e] * dinv[cnode];
    const float4 v  = ((const float4*)(T + (size_t)r * H_DIM))[lane];
    float* __restrict__ dst = agg + (size_t)cnode * H_DIM + (lane << 2);
    atomicAdd(dst + 0, nrm * v.x);
    atomicAdd(dst + 1, nrm * v.y);
    atomicAdd(dst + 2, nrm * v.z);
    atomicAdd(dst + 3, nrm * v.w);
}

// ---------------- bias + ReLU + BN statistics ----------------
__global__ void k_zero(float* __restrict__ p, int n) {
    int i = blockIdx.x * blockDim.x + threadIdx.x;
    if (i < n) p[i] = 0.f;
}

// 128 threads <-> 128 features; per-block partial reduction over a node chunk,
// then 2 atomics per feature per block into sums[0:128]=sum, sums[128:256]=sumsq.
__global__ __launch_bounds__(128) void k_bias_relu_stats(
    const float* __restrict__ agg, const float* __restrict__ bias,
    float* __restrict__ Y, float* __restrict__ sums, int n, int npb)
{
    const int f  = threadIdx.x;
    const float bf = bias[f];
    const int n0 = blockIdx.x * npb;
    const int n1 = min(n0 + npb, n);
    float s = 0.f, s2 = 0.f;
    for (int node = n0; node < n1; ++node) {
        __builtin_prefetch(agg + (size_t)(node + 16) * H_DIM + f, 0, 1); // global_prefetch_b8
        float v = agg[(size_t)node * H_DIM + f] + bf;
        v = fmaxf(v, 0.f);
        Y[(size_t)node * H_DIM + f] = v;
        s += v; s2 += v * v;
    }
    atomicAdd(&sums[f], s);
    atomicAdd(&sums[H_DIM + f], s2);
}

__global__ void k_bn_coeffs(const float* __restrict__ sums,
                            const float* __restrict__ gamma, const float* __restrict__ beta,
                            float* __restrict__ scale, float* __restrict__ shift, int n)
{
    const int f = threadIdx.x;
    if (f < H_DIM) {
        const float inv_n = 1.0f / (float)n;
        const float mu  = sums[f] * inv_n;
        const float var = sums[H_DIM + f] * inv_n - mu * mu;   // biased variance
        const float sc  = gamma[f] * rsqrtf(var + BN_EPS);
        scale[f] = sc;
        shift[f] = beta[f] - sc * mu;
    }
}

// ---------------- final linear (128 -> 10), BN affine fused into the load ----------------
__global__ __launch_bounds__(256) void k_final(
    const float* __restrict__ Y, const float* __restrict__ scale, const float* __restrict__ shift,
    const float* __restrict__ Wl, const float* __restrict__ bl,
    float* __restrict__ out, int n, int ncls)
{
    const int idx  = blockIdx.x * blockDim.x + threadIdx.x;
    const int node = idx >> 4;
    const int c    = idx & 15;
    if (node >= n || c >= ncls) return;
    float acc = bl[c];
    const float* __restrict__ yr = Y + (size_t)node * H_DIM;
    #pragma unroll 8
    for (int k = 0; k < H_DIM; ++k) {
        const float h = yr[k] * scale[k] + shift[k];
        acc += h * Wl[k * ncls + c];
    }
    out[(size_t)node * ncls + c] = acc;
}

// ---------------- host orchestration ----------------
extern "C" void kernel_launch(void* const* d_in, const int* in_sizes, int n_in,
                              void* d_out, int out_size, void* d_ws, size_t ws_size,
                              hipStream_t stream)
{
    const int N = in_sizes[0] / H_DIM;       // 100000
    const int E = in_sizes[2];               // 1600000
    const int C = in_sizes[16];              // 10 (bl)

    const float* x    = (const float*)d_in[0];
    const int*   eidx = (const int*)  d_in[1];
    const float* ew   = (const float*)d_in[2];
    const float* Wmat[3] = { (const float*)d_in[3], (const float*)d_in[5], (const float*)d_in[7] };
    const float* bvec[3] = { (const float*)d_in[4], (const float*)d_in[6], (const float*)d_in[8] };
    const float* gvec[3] = { (const float*)d_in[9],  (const float*)d_in[11], (const float*)d_in[13] };
    const float* bevec[3]= { (const float*)d_in[10], (const float*)d_in[12], (const float*)d_in[14] };
    const float* Wl = (const float*)d_in[15];
    const float* bl = (const float*)d_in[16];
    float* out = (float*)d_out;

    const int* rows = eidx;          // edge_index[0]
    const int* cols = eidx + E;      // edge_index[1]

    // workspace layout (2 ping-pong feature buffers + small scalars)
    const size_t NB = (size_t)N * H_DIM;     // elements
    float* bufA  = (float*)d_ws;             // 51.2 MB
    float* bufB  = bufA + NB;                // 51.2 MB
    float* deg   = bufB + NB;                // N
    float* dinv  = deg + N;                  // N
    float* sums  = dinv + N;                 // 256
    float* scl   = sums + 256;               // 128
    float* shf   = scl + 128;                // 128
    (void)ws_size; (void)n_in; (void)out_size;

    const dim3 b256(256), b128(128);
    const unsigned gN    = (N + 255) / 256;
    const unsigned gE    = (E + 255) / 256;
    const unsigned gNH   = ((unsigned)NB + 255) / 256;
    const unsigned gSc   = (E + 7) / 8;
    const unsigned gGemm = (N + 127) / 128;          // 128 rows per block (8 waves x 16)
    const unsigned statBlocks = 512;
    const int npb = (N + (int)statBlocks - 1) / (int)statBlocks;

    // --- normalization ---
    k_deg_init <<<gN, b256, 0, stream>>>(deg, N);
    k_deg_edges<<<gE, b256, 0, stream>>>(cols, ew, deg, E);
    k_dinv     <<<gN, b256, 0, stream>>>(deg, dinv, N);

    const float* in = x;
    float* T   = bufA;
    float* AGG = bufB;

    for (int layer = 0; layer < 3; ++layer) {
        // T = (in [*scale+shift]) @ W   [WMMA + TDM-staged W]
        if (layer == 0)
            k_gemm_wmma<false><<<gGemm, b256, 0, stream>>>(in, Wmat[layer], nullptr, nullptr, T, N);
        else
            k_gemm_wmma<true ><<<gGemm, b256, 0, stream>>>(in, Wmat[layer], scl, shf, T, N);
        // AGG = selfnorm*T, then scatter-add edge messages
        k_agg_init<<<gNH, b256, 0, stream>>>(T, dinv, AGG, N);
        k_scatter <<<gSc, b256, 0, stream>>>(rows, cols, ew, dinv, T, AGG, E);
        // Y = relu(AGG + b)  (written over T's buffer) + batch statistics
        k_zero<<<1, b256, 0, stream>>>(sums, 256);
        k_bias_relu_stats<<<statBlocks, b128, 0, stream>>>(AGG, bvec[layer], T, sums, N, npb);
        k_bn_coeffs<<<1, b128, 0, stream>>>(sums, gvec[layer], bevec[layer], scl, shf, N);
        // next layer reads Y (= T buffer) with BN affine fused into GEMM load
        in = T;
        float* tmp = T; T = AGG; AGG = tmp;   // ping-pong
    }

    // out = BN(Y3) @ Wl + bl ; Y3 is `in`
    const unsigned gF = ((unsigned)N * 16 + 255) / 256;
    k_final<<<gF, b256, 0, stream>>>(in, scl, shf, Wl, bl, out, N, C);
}